// ImageReconstruction_35399120453774
// MI455X (gfx1250) — compile-verified
//
#include <hip/hip_runtime.h>

typedef __attribute__((ext_vector_type(2))) float v2f;
typedef __attribute__((ext_vector_type(8))) float v8f;

constexpr int kB    = 16;      // batch
constexpr int kH    = 512;
constexpr int kW    = 512;
constexpr int kBL   = 64;      // block_len (features per patch)
constexpr int kNB   = 256;     // codebook entries
constexpr int kL    = 4096;    // patches per image
constexpr float kBeta = 35.0f;
constexpr int kWSTR = 264;     // padded LDS row stride (floats)
constexpr int kNWG  = (kL / 16) * kB;   // 4096 workgroups

// workspace layout (floats)
constexpr int kWsNorm = 0;               // [256]      codebook squared norms
constexpr int kWsCbT  = kNB;             // [64*256]   transposed codebook cbT[f][n]
constexpr int kWsPart = kNB + kBL * kNB; // [4096]     penalty partials

// ---------------------------------------------------------------------------
// Kernel 1: codebook squared norms + transpose.
//   ws[n]              = sum_k cb[n][k]^2
//   cbT[k*256 + n]     = cb[n*64 + k]        (B operand for GEMM2, L2-resident)
// ---------------------------------------------------------------------------
__global__ __launch_bounds__(256) void cb_prep_kernel(const float* __restrict__ cb,
                                                      float* __restrict__ ws) {
    int n = threadIdx.x;
    const float4* p = (const float4*)(cb + n * kBL);
    float* cbT = ws + kWsCbT;
    float s = 0.f;
#pragma unroll
    for (int i = 0; i < 16; ++i) {
        float4 v = p[i];
        s += v.x * v.x + v.y * v.y + v.z * v.z + v.w * v.w;
        cbT[(4 * i + 0) * kNB + n] = v.x;
        cbT[(4 * i + 1) * kNB + n] = v.y;
        cbT[(4 * i + 2) * kNB + n] = v.z;
        cbT[(4 * i + 3) * kNB + n] = v.w;
    }
    ws[kWsNorm + n] = s;
}

// ---------------------------------------------------------------------------
// Kernel 2: main reconstruction. One wave32 per (image, 16-patch tile).
//   grid = (256, 16), block = 32
// ---------------------------------------------------------------------------
__global__ __launch_bounds__(32) void recon_kernel(const float* __restrict__ img,
                                                   const float* __restrict__ cb,
                                                   float* __restrict__ out,
                                                   float* __restrict__ ws) {
    __shared__ float a2s[16];
    __shared__ float b2s[kNB];
    __shared__ float wlds[16 * kWSTR];   // scores -> exp -> weights, [patch][code]

    const int lane = threadIdx.x;        // 0..31 (wave32)
    const int mrow = lane & 15;
    const int hl   = lane >> 4;          // half-wave id
    const int b    = blockIdx.y;
    const int l0   = blockIdx.x * 16;    // first patch index of this tile
    const int hb   = l0 >> 6;            // patch-grid row (constant across tile)
    const int wb0  = l0 & 63;            // patch-grid col base

    const float* cbT = ws + kWsCbT;

    // warm the L2 path for both codebook copies (64 KB each, 2 KB per lane)
    __builtin_prefetch(cb + lane * 512, 0, 0);
    __builtin_prefetch(cbT + lane * 512, 0, 0);

    // codebook norms from workspace -> LDS
#pragma unroll
    for (int i = 0; i < 8; ++i) b2s[lane * 8 + i] = ws[kWsNorm + lane * 8 + i];

    // ---- load A fragments (uf tile, 16 patches x 64 feats) + uf norms ----
    // f32 16x4 A layout: lanes 0-15 hold (M=lane, K=4kc+{0,1}),
    //                    lanes 16-31 hold (M=lane-16, K=4kc+{2,3})
    const float* ibase = img + b * (kH * kW);
    const float* pbase = ibase + (hb * 8) * kW + (wb0 + mrow) * 8;
    v2f  afrag[16];
    float a2p = 0.f;
#pragma unroll
    for (int kc = 0; kc < 16; ++kc) {
        int k0 = 4 * kc + 2 * hl;                       // even -> same pixel row
        v2f v = *(const v2f*)(pbase + (k0 >> 3) * kW + (k0 & 7));
        afrag[kc] = v;
        a2p += v.x * v.x + v.y * v.y;
    }
    float a2row = a2p + __shfl_xor(a2p, 16, 32);        // combine K-halves
    if (hl == 0) a2s[mrow] = a2row;
    __syncthreads();

    float a2v[8];
#pragma unroll
    for (int r = 0; r < 8; ++r) a2v[r] = a2s[r + 8 * hl];

    // ---- GEMM1: dist[16 x 256] = ||uf||^2 + ||cb||^2 - 2 * uf . cb ----
    for (int t = 0; t < 16; ++t) {
        const int n = 16 * t + mrow;                    // this lane's code column
        const float* crow = cb + n * kBL + 2 * hl;      // B frag: cb[n][4kc+2hl..+1]
        v8f acc = {};
#pragma unroll
        for (int kc = 0; kc < 16; ++kc) {
            v2f bf = *(const v2f*)(crow + 4 * kc);
            acc = __builtin_amdgcn_wmma_f32_16x16x4_f32(
                false, afrag[kc], false, bf, (short)0, acc, false, false);
        }
        float b2 = b2s[n];
#pragma unroll
        for (int r = 0; r < 8; ++r) {                   // C layout: (M=r+8*hl, N=n)
            int m = r + 8 * hl;
            float s = -kBeta * (a2v[r] + b2 - 2.0f * acc[r]);
            wlds[m * kWSTR + n] = s;
        }
    }
    __syncthreads();

    // ---- softmax over 256 codes per patch (2 lanes per row) + penalty ----
    {
        const int row = lane >> 1;
        const int c0  = (lane & 1) * 128;
        float* wr = &wlds[row * kWSTR + c0];
        float mx = -3.0e38f;
#pragma unroll 8
        for (int j = 0; j < 128; ++j) mx = fmaxf(mx, wr[j]);
        mx = fmaxf(mx, __shfl_xor(mx, 1, 32));
        float sum = 0.f;
#pragma unroll 8
        for (int j = 0; j < 128; ++j) {
            float e = __expf(wr[j] - mx);
            wr[j] = e;
            sum += e;
        }
        sum += __shfl_xor(sum, 1, 32);
        float inv = 1.0f / sum;
        float pen = 0.f;
#pragma unroll 8
        for (int j = 0; j < 128; ++j) {
            float w = wr[j] * inv;
            wr[j] = w;
            pen += fminf(w, 1.0f - w);
        }
#pragma unroll
        for (int off = 16; off >= 1; off >>= 1) pen += __shfl_xor(pen, off, 32);
        if (lane == 0) ws[kWsPart + blockIdx.y * gridDim.x + blockIdx.x] = pen;
    }
    __syncthreads();

    // ---- GEMM2: OUT[16 x 64] = W[16 x 256] x CB[256 x 64], fold to image ----
    // B operand read from cbT[f][n]: pairs along the code index are contiguous
    // -> one aligned global_load_b64 per WMMA (same shape as GEMM1).
    float* obase = out + b * (kH * kW);
    const float* arow = &wlds[mrow * kWSTR + 2 * hl];
#pragma unroll
    for (int tn = 0; tn < 4; ++tn) {
        const int f = 16 * tn + mrow;                   // this lane's feature column
        const float* btrow = cbT + f * kNB + 2 * hl;    // cbT[f][4kc+2hl..+1]
        v8f acc = {};
#pragma unroll
        for (int kc = 0; kc < 64; ++kc) {
            v2f a  = *(const v2f*)(arow + 4 * kc);       // W[mrow][kk..kk+1]
            v2f bf = *(const v2f*)(btrow + 4 * kc);      // cb[kk..kk+1][f]
            acc = __builtin_amdgcn_wmma_f32_16x16x4_f32(
                false, a, false, bf, (short)0, acc, false, false);
        }
        const int prow  = hb * 8 + (f >> 3);
        const int pcol0 = wb0 * 8 + (f & 7);
#pragma unroll
        for (int r = 0; r < 8; ++r) {                    // (patch = r+8*hl, feat = f)
            int p = r + 8 * hl;
            obase[prow * kW + pcol0 + p * 8] = acc[r];
        }
    }
}

// ---------------------------------------------------------------------------
// Kernel 3: deterministic fixed-tree reduction of the 4096 penalty partials.
//   out[B*H*W] = (sum partials) / (B * L)
// ---------------------------------------------------------------------------
__global__ __launch_bounds__(256) void pen_reduce_kernel(const float* __restrict__ ws,
                                                         float* __restrict__ out) {
    __shared__ float sm[256];
    int tid = threadIdx.x;
    float s = 0.f;
    for (int i = tid; i < kNWG; i += 256) s += ws[kWsPart + i];  // fixed order
    sm[tid] = s;
    __syncthreads();
    for (int off = 128; off > 0; off >>= 1) {
        if (tid < off) sm[tid] += sm[tid + off];
        __syncthreads();
    }
    if (tid == 0) out[kB * kH * kW] = sm[0] * (1.0f / (float)(kB * kL));
}

extern "C" void kernel_launch(void* const* d_in, const int* in_sizes, int n_in,
                              void* d_out, int out_size, void* d_ws, size_t ws_size,
                              hipStream_t stream) {
    const float* img = (const float*)d_in[0];   // [16,1,512,512] f32
    const float* cb  = (const float*)d_in[1];   // [256,8,8] f32
    float* out = (float*)d_out;                  // [16*512*512] + 1 penalty scalar
    float* ws  = (float*)d_ws;                   // norms + cbT + partials (~83 KB)

    cb_prep_kernel<<<1, 256, 0, stream>>>(cb, ws);
    dim3 grid(kL / 16, kB);
    recon_kernel<<<grid, 32, 0, stream>>>(img, cb, out, ws);
    pen_reduce_kernel<<<1, 256, 0, stream>>>(ws, out);
}